// Maxk_Pooling_Variable_5712306503682
// MI455X (gfx1250) — compile-verified
//
#include <hip/hip_runtime.h>
#include <hip/hip_bf16.h>

// Problem constants (from the reference): features [B, L, D] f32, lengths [B] int
#define BATCH     32
#define SEQLEN    4096
#define DIM       512
#define KTOP      8
#define TILE_ROWS 16           // rows per TDM tile (16 x 512 x 4B = 32KB)

typedef unsigned int u32x4 __attribute__((ext_vector_type(4)));
typedef int          i32x8 __attribute__((ext_vector_type(8)));
typedef int          i32x4 __attribute__((ext_vector_type(4)));

#if defined(__gfx1250__) && __has_builtin(__builtin_amdgcn_tensor_load_to_lds) && \
    __has_builtin(__builtin_amdgcn_s_wait_tensorcnt)
#define USE_TDM 1
#else
#define USE_TDM 0
#endif

// Sorted (descending) top-K insertion; t[] stays sorted so pass2 can take a
// prefix directly (reference takes the first k_eff of the sorted top-k list).
__device__ __forceinline__ void topk_insert(float (&t)[KTOP], float v) {
    if (v > t[KTOP - 1]) {
        t[KTOP - 1] = v;
#pragma unroll
        for (int j = KTOP - 1; j > 0; --j) {
            if (t[j] > t[j - 1]) {
                float tmp = t[j - 1];
                t[j - 1]  = t[j];
                t[j]      = tmp;
            }
        }
    }
}

#if USE_TDM
// Issue one TDM tile load: TILE_ROWS x DIM f32, row-contiguous, global -> LDS.
// D# packing per CDNA5 ISA ch.8 (group0: count/lds/global/type; group1: dims).
// This toolchain exposes the 6-arg builtin:
//   (u32x4 g0, i32x8 g1, i32x4 g2, i32x4 g3, i32x8 extra, i32 cpol)
__device__ __forceinline__ void tdm_load_tile(const float* gsrc, unsigned lds_byte_off) {
    unsigned long long ga = (unsigned long long)(uintptr_t)gsrc;

    u32x4 g0;
    g0.s0 = 1u;                                            // count=1, user desc
    g0.s1 = lds_byte_off;                                  // lds_addr (bytes)
    g0.s2 = (unsigned)(ga & 0xFFFFFFFFull);                // global_addr[31:0]
    g0.s3 = (unsigned)((ga >> 32) & 0x1FFFFFFull)          // global_addr[56:32]
          | (2u << 30);                                    // type = 2 ("image")

    i32x8 g1;
    g1.s0 = (int)(2u << 16);                               // data_size = 4 bytes
    g1.s1 = (int)((unsigned)DIM << 16);                    // tensor_dim0[15:0] = 512
    g1.s2 = (int)((unsigned)TILE_ROWS << 16);              // tensor_dim1[15:0]
    g1.s3 = (int)((unsigned)DIM << 16);                    // tile_dim0 = 512
    g1.s4 = TILE_ROWS;                                     // tile_dim1, tile_dim2 = 0
    g1.s5 = DIM;                                           // tensor_dim0_stride = 512
    g1.s6 = 0;
    g1.s7 = 0;

    i32x4 z4 = {0, 0, 0, 0};                               // groups 2/3: 2-D tensor
    i32x8 z8 = {0, 0, 0, 0, 0, 0, 0, 0};
    __builtin_amdgcn_tensor_load_to_lds(g0, g1, z4, z4, z8, 0);
}
#endif

// ---------------- Pass 1: per (b, segment) partial top-8 per column ----------
__global__ void __launch_bounds__(DIM)
maxk_pass1(const float* __restrict__ feat, const int* __restrict__ lengths,
           float* __restrict__ ws, int nseg, int seg_rows) {
    const int b    = blockIdx.x;
    const int seg  = blockIdx.y;
    const int d    = threadIdx.x;
    const int len  = lengths[b];
    const int row0 = seg * seg_rows;

    float t[KTOP];
#pragma unroll
    for (int j = 0; j < KTOP; ++j) t[j] = -__builtin_inff();

    float* out = ws + (((size_t)b * nseg + seg) * DIM + d) * KTOP;

    // Entirely masked segment: no DMA at all (this halves average HBM traffic
    // since lengths ~ U[8, 4096]). Uniform per-block branch.
    const int valid = len - row0;
    if (valid <= 0) {
#pragma unroll
        for (int j = 0; j < KTOP; ++j) out[j] = t[j];
        return;
    }

    const float* base = feat + ((size_t)b * SEQLEN + (size_t)row0) * DIM;
    int ntiles = (valid + TILE_ROWS - 1) / TILE_ROWS;     // clamp to valid rows
    const int cap = seg_rows / TILE_ROWS;
    if (ntiles > cap) ntiles = cap;

#if USE_TDM
    extern __shared__ float lds[];
    const unsigned tile_bytes = TILE_ROWS * DIM * 4u;

    if (threadIdx.x == 0) {
        tdm_load_tile(base, 0u);
        if (ntiles > 1) tdm_load_tile(base + (size_t)TILE_ROWS * DIM, tile_bytes);
    }
    for (int tl = 0; tl < ntiles; ++tl) {
        if (threadIdx.x == 0) {
            if (tl + 1 < ntiles) __builtin_amdgcn_s_wait_tensorcnt(1);
            else                 __builtin_amdgcn_s_wait_tensorcnt(0);
        }
        __syncthreads();                      // tile tl resident in lds[tl&1]
        const float* buf = lds + (size_t)(tl & 1) * (TILE_ROWS * DIM);
        const int lbase  = row0 + tl * TILE_ROWS;
        if (lbase + TILE_ROWS <= len) {       // fully valid tile: no mask test
#pragma unroll
            for (int r = 0; r < TILE_ROWS; ++r)
                topk_insert(t, buf[r * DIM + d]);
        } else {                              // tail tile: per-row mask
#pragma unroll
            for (int r = 0; r < TILE_ROWS; ++r) {
                float v = buf[r * DIM + d];
                if (lbase + r < len) topk_insert(t, v);
            }
        }
        __syncthreads();                      // everyone done with lds[tl&1]
        const int nxt = tl + 2;
        if (nxt < ntiles && threadIdx.x == 0) {
            tdm_load_tile(base + (size_t)nxt * TILE_ROWS * DIM,
                          (unsigned)(nxt & 1) * tile_bytes);
        }
    }
#else
    const int rows = (valid < seg_rows) ? valid : seg_rows;
    for (int r = 0; r < rows; ++r)
        topk_insert(t, base[(size_t)r * DIM + d]);
#endif

#pragma unroll
    for (int j = 0; j < KTOP; ++j) out[j] = t[j];
}

// ---------------- Pass 2: merge segments, cummin k_eff, mean -----------------
__global__ void maxk_pass2(const float* __restrict__ ws,
                           const int* __restrict__ lengths,
                           float* __restrict__ out, int nseg) {
    int idx = blockIdx.x * blockDim.x + threadIdx.x;
    if (idx >= BATCH * DIM) return;
    int b = idx / DIM;
    int d = idx - b * DIM;

    float t[KTOP];
#pragma unroll
    for (int j = 0; j < KTOP; ++j) t[j] = -__builtin_inff();

    for (int s = 0; s < nseg; ++s) {
        const float* c = ws + (((size_t)b * nseg + s) * DIM + d) * KTOP;
#pragma unroll
        for (int j = 0; j < KTOP; ++j) topk_insert(t, c[j]);
    }

    // k_eff = min(K, cummin(lengths)[b])  (B is tiny; loads hit cache)
    int m = lengths[0];
    for (int i = 1; i <= b; ++i) { int li = lengths[i]; m = (li < m) ? li : m; }
    int keff = (m < KTOP) ? m : KTOP;

    float sum = 0.0f;
#pragma unroll
    for (int j = 0; j < KTOP; ++j)
        if (j < keff) sum += t[j];
    out[idx] = sum / (float)keff;
}

// ---------------- Fused fallback (workspace too small) -----------------------
__global__ void maxk_fused(const float* __restrict__ feat,
                           const int* __restrict__ lengths,
                           float* __restrict__ out) {
    int idx = blockIdx.x * blockDim.x + threadIdx.x;
    if (idx >= BATCH * DIM) return;
    int b = idx / DIM;
    int d = idx - b * DIM;
    int len = lengths[b];
    const float* col = feat + (size_t)b * SEQLEN * DIM + d;

    float t[KTOP];
#pragma unroll
    for (int j = 0; j < KTOP; ++j) t[j] = -__builtin_inff();
    for (int l = 0; l < len; ++l) topk_insert(t, col[(size_t)l * DIM]);

    int m = lengths[0];
    for (int i = 1; i <= b; ++i) { int li = lengths[i]; m = (li < m) ? li : m; }
    int keff = (m < KTOP) ? m : KTOP;

    float sum = 0.0f;
#pragma unroll
    for (int j = 0; j < KTOP; ++j)
        if (j < keff) sum += t[j];
    out[idx] = sum / (float)keff;
}

extern "C" void kernel_launch(void* const* d_in, const int* in_sizes, int n_in,
                              void* d_out, int out_size, void* d_ws, size_t ws_size,
                              hipStream_t stream) {
    (void)in_sizes; (void)n_in; (void)out_size;
    const float* feat    = (const float*)d_in[0];
    const int*   lengths = (const int*)d_in[1];
    float*       out     = (float*)d_out;

    const size_t per_seg = (size_t)BATCH * DIM * KTOP * sizeof(float); // 512KB
    int nseg = 16;
    while (nseg > 1 && per_seg * (size_t)nseg > ws_size) nseg >>= 1;

    const int n = BATCH * DIM;
    if (per_seg > ws_size) {
        // No usable workspace: fused single-pass kernel.
        maxk_fused<<<(n + 255) / 256, 256, 0, stream>>>(feat, lengths, out);
        return;
    }

    const int seg_rows = SEQLEN / nseg;                   // powers of two: exact
    dim3 grid1(BATCH, nseg);
    size_t shmem = 2u * TILE_ROWS * DIM * sizeof(float);  // 64KB double buffer
    maxk_pass1<<<grid1, DIM, shmem, stream>>>(feat, lengths, (float*)d_ws,
                                              nseg, seg_rows);
    maxk_pass2<<<(n + 255) / 256, 256, 0, stream>>>((const float*)d_ws, lengths,
                                                    out, nseg);
}